// GroupWiseCorr_5D_67370857005289
// MI455X (gfx1250) — compile-verified
//
#include <hip/hip_runtime.h>
#include <cstdint>

namespace {
constexpr int kN  = 2;
constexpr int kS  = 16;
constexpr int kC  = 64;
constexpr int kH  = 128;
constexpr int kW  = 160;
constexpr int kG  = 8;               // NUM_GROUPS
constexpr int kHW  = kH * kW;        // 20480
constexpr int kCHW = kC * kHW;
constexpr int kWT  = 40;             // w-tile per block
constexpr int kTiles  = kW / kWT;    // 4
constexpr int kChunks = kC * kWT / 4; // 640 x b128 chunks for the LDS ref tile

typedef float v2f_t __attribute__((ext_vector_type(2)));
typedef float v8f_t __attribute__((ext_vector_type(8)));
} // namespace

__global__ __launch_bounds__(256)
void gwc5d_wmma_kernel(const float* __restrict__ feat_ref,
                       const float* __restrict__ feat_src,
                       const float* __restrict__ grids,
                       const float* __restrict__ mask,
                       float* __restrict__ out_corr,
                       float* __restrict__ out_wm)
{
  __shared__ float ref_tile[kC * kWT]; // 10 KB: feat_ref[n, 0:64, h, w0:w0+40]

  const int blk = (int)blockIdx.x;
  const int wt  = blk % kTiles;
  const int h   = (blk / kTiles) % kH;
  const int n   = blk / (kTiles * kH);
  const int w0  = wt * kWT;
  const int tid = (int)threadIdx.x;

  // ---- Stage the reference tile into LDS with CDNA5 async global->LDS copies ----
  {
    const float* gbase = feat_ref + (size_t)n * kCHW + (size_t)h * kW + w0;
    // addrspace(3) flat addresses carry the LDS byte offset in the low 32 bits
    const uint32_t lds_base = (uint32_t)(uintptr_t)(&ref_tile[0]);
    for (int q = tid; q < kChunks; q += 256) {
      const int c = q / (kWT / 4);
      const int t = q % (kWT / 4);
      const uint32_t voff = (uint32_t)(c * kHW + t * 4) * 4u; // byte offset vs gbase
      const uint32_t ldsa = lds_base + (uint32_t)q * 16u;
      asm volatile("global_load_async_to_lds_b128 %0, %1, %2"
                   :: "v"(ldsa), "v"(voff), "s"(gbase)
                   : "memory");
    }
    asm volatile("s_wait_asynccnt 0" ::: "memory");
    __syncthreads();
  }

  const int lane = tid & 31;
  const int wave = tid >> 5;
  const int sN   = lane & 15;  // A-row (sample s) and B/D-column (group g)
  const int half = lane >> 4;  // which channel pair of each K=4 slice this lane owns

  const float Wf = (float)kW, Hf = (float)kH;

  for (int wl = wave; wl < kWT; wl += 8) {
    const int w = w0 + wl;

    // grid coords for (n, s=sN, h, w); lanes L and L+16 duplicate the same s
    const int gidx = (((n * kS + sN) * 2) * kH + h) * kW + w;
    const float xg = grids[gidx];
    const float yg = grids[gidx + kHW];

    // replicate the reference's exact f32 arithmetic
    const float xn = 2.0f * xg / Wf - 1.0f;
    const float yn = 2.0f * yg / Hf - 1.0f;
    const float gx = ((xn + 1.0f) * Wf - 1.0f) / 2.0f;
    const float gy = ((yn + 1.0f) * Hf - 1.0f) / 2.0f;

    const float x0f = floorf(gx), y0f = floorf(gy);
    const float dx = gx - x0f, dy = gy - y0f;
    const int x0 = (int)x0f, y0 = (int)y0f;
    const int x1 = x0 + 1,  y1 = y0 + 1;

    const float vx0 = (x0 >= 0 && x0 < kW) ? 1.0f : 0.0f;
    const float vx1 = (x1 >= 0 && x1 < kW) ? 1.0f : 0.0f;
    const float vy0 = (y0 >= 0 && y0 < kH) ? 1.0f : 0.0f;
    const float vy1 = (y1 >= 0 && y1 < kH) ? 1.0f : 0.0f;

    const float wvA = ((1.0f - dx) * (1.0f - dy)) * (vx0 * vy0);
    const float wvB = (dx * (1.0f - dy)) * (vx1 * vy0);
    const float wvC = ((1.0f - dx) * dy) * (vx0 * vy1);
    const float wvD = (dx * dy) * (vx1 * vy1);
    const float m   = ((wvA + wvB) + wvC) + wvD;
    const float wmask = (m < 0.9999f) ? 0.0f : 1.0f;
    const float wm = wmask * mask[((n * kS + sN) * kH + h) * kW + w];

    const int x0c = x0 < 0 ? 0 : (x0 > kW - 1 ? kW - 1 : x0);
    const int x1c = x1 < 0 ? 0 : (x1 > kW - 1 ? kW - 1 : x1);
    const int y0c = y0 < 0 ? 0 : (y0 > kH - 1 ? kH - 1 : y0);
    const int y1c = y1 < 0 ? 0 : (y1 > kH - 1 ? kH - 1 : y1);
    const int oA = y0c * kW + x0c;
    const int oB = y0c * kW + x1c;
    const int oC = y1c * kW + x0c;
    const int oD = y1c * kW + x1c;

    // Bilinear gather of this lane's 32 channels -> WMMA A fragments.
    // ISA 16x4 f32 A layout: lane L (M=L%16): VGPR0=K(0|2), VGPR1=K(1|3) by lane half.
    const float* srcp = feat_src + (size_t)n * kCHW + (size_t)(2 * half) * kHW;
    v2f_t a[16];
#pragma unroll
    for (int k = 0; k < 16; ++k) {
      const float* p0 = srcp + (size_t)(4 * k) * kHW;
      const float* p1 = p0 + kHW;
      a[k].x = ((p0[oA] * wvA + p0[oB] * wvB) + p0[oC] * wvC) + p0[oD] * wvD;
      a[k].y = ((p1[oA] * wvA + p1[oB] * wvB) + p1[oC] * wvC) + p1[oD] * wvD;
    }

    // Block-diagonal ref matrix (B[c,g] = ref[c] iff g == c/8), contracted with
    // 16 chained f32 WMMAs: corr[s,g] = sum_c warped[s,c] * ref[c] * [g==c/8].
    v8f_t acc = {};
#pragma unroll
    for (int k = 0; k < 16; ++k) {
      const int c0 = 4 * k + 2 * half;
      const float r0 = ref_tile[c0 * kWT + wl];       // LDS broadcast reads
      const float r1 = ref_tile[(c0 + 1) * kWT + wl];
      const bool sel = (sN == (k >> 1));              // group of channels 4k..4k+3
      v2f_t b;
      b.x = sel ? r0 : 0.0f;
      b.y = sel ? r1 : 0.0f;
      acc = __builtin_amdgcn_wmma_f32_16x16x4_f32(false, a[k], false, b,
                                                  (short)0, acc, false, false);
    }

    if (lane < 16) {
      out_wm[((n * kS + sN) * kH + h) * kW + w] = wm;
    }
    // D layout: lane L, vgpr j -> (s = j + 8*(L/16), g = L%16)
#pragma unroll
    for (int j = 0; j < 8; ++j) {
      const int sj = j + 8 * half;
      const float wmj = __shfl(wm, sj, 32); // lane sj (0..15) holds wm for sample sj
      if (sN < kG) {
        out_corr[(((size_t)(n * kG + sN) * kS + sj) * kH + h) * kW + w] = acc[j] * wmj;
      }
    }
  }
}

extern "C" void kernel_launch(void* const* d_in, const int* in_sizes, int n_in,
                              void* d_out, int out_size, void* d_ws, size_t ws_size,
                              hipStream_t stream) {
  (void)in_sizes; (void)n_in; (void)out_size; (void)d_ws; (void)ws_size;
  const float* feat_ref = (const float*)d_in[0];
  const float* feat_src = (const float*)d_in[1];
  const float* grids    = (const float*)d_in[2];
  const float* mask     = (const float*)d_in[3];
  float* out_corr = (float*)d_out;
  float* out_wm   = (float*)d_out + (size_t)kN * kG * kS * kH * kW;

  dim3 grid(kN * kH * kTiles); // 1024 blocks
  dim3 block(256);             // 8 wave32s
  gwc5d_wmma_kernel<<<grid, block, 0, stream>>>(feat_ref, feat_src, grids, mask,
                                                out_corr, out_wm);
}